// RYGate_549755814189
// MI455X (gfx1250) — compile-verified
//
#include <hip/hip_runtime.h>

typedef __attribute__((ext_vector_type(2))) float v2f;
typedef __attribute__((ext_vector_type(4))) float v4f;
typedef __attribute__((ext_vector_type(8))) float v8f;

#define NQ  13
#define DIM 8192
#define NT  512

// m = [[c, s], [-s, c]] : one 2x2 truncated-RY factor
__device__ __forceinline__ float gate_elem(float c, float s, int rb, int kb) {
    return rb == 0 ? (kb == 0 ? c : s) : (kb == 0 ? -s : c);
}

// State index for (row-in-group k, column col) for each WMMA pass.
// pass0: group bits 12-9  -> idx = k*512 + col                (col = bits 8-0)
// pass1: group bits 8-5   -> idx = o*512 + k*32 + i           (col = o*32 + i)
// pass2: group bits 4-1   -> idx = o*32 + k*2 + i             (col = o*2 + i)
__device__ __forceinline__ int paddr(int pass, int k, int col) {
    if (pass == 0) return (k << 9) + col;
    if (pass == 1) return ((col >> 5) << 9) + (k << 5) + (col & 31);
    return ((col >> 1) << 5) + (k << 1) + (col & 1);
}

__global__ __launch_bounds__(NT) void ry13_wmma_kernel(
    const float* __restrict__ xr, const float* __restrict__ xi,
    const float* __restrict__ ang, float* __restrict__ out)
{
    __shared__ float st[DIM];   // one full component (real or imag) of the state
    __shared__ float gc[NQ];    // truncated cos(theta/2) per site
    __shared__ float gs[NQ];    // truncated sin(theta/2) per site

    const int tid  = threadIdx.x;
    const int comp = blockIdx.x;            // 0 = real part, 1 = imag part
    const float* __restrict__ x = (comp == 0) ? xr : xi;

    // ---- per-site gate coefficients (cutoff-10 power series, matches reference) ----
    if (tid < NQ) {
        float t  = 0.5f * ang[tid];
        float t2 = t * t;
        float s = t * (1.0f + t2*(-1.0f/6.0f + t2*(1.0f/120.0f
                  + t2*(-1.0f/5040.0f + t2*(1.0f/362880.0f)))));
        float c = 1.0f + t2*(-0.5f + t2*(1.0f/24.0f + t2*(-1.0f/720.0f
                  + t2*(1.0f/40320.0f + t2*(-1.0f/3628800.0f)))));
        gc[tid] = c; gs[tid] = s;
    }

    // ---- load this component's 8192 floats into LDS (coalesced b128) ----
    {
        const v4f* xv = (const v4f*)x;
        v4f* sv = (v4f*)st;
        #pragma unroll
        for (int i = 0; i < DIM/4/NT; ++i) sv[tid + i*NT] = xv[tid + i*NT];
    }
    __syncthreads();

    const int lane = tid & 31;
    const int hi   = lane >> 4;   // lane half (K/M split per ISA fragment layout)
    const int ln   = lane & 15;   // M for A-frag, N (column) for B/D frags
    const int wave = tid >> 5;    // 0..15

    // ---- three passes of 16x16 kron-group matrices via f32 WMMA ----
    #pragma unroll
    for (int pass = 0; pass < 3; ++pass) {
        const int sb = pass * 4;  // sites sb..sb+3 -> bits (12-sb)..(9-sb)
        const float c0 = gc[sb+0], s0 = gs[sb+0];
        const float c1 = gc[sb+1], s1 = gs[sb+1];
        const float c2 = gc[sb+2], s2 = gs[sb+2];
        const float c3 = gc[sb+3], s3 = gs[sb+3];

        // A fragments: chunk j holds G[:, 4j..4j+3]; lane l: M=ln, K=4j+v+2*hi
        v2f a[4];
        #pragma unroll
        for (int j = 0; j < 4; ++j) {
            #pragma unroll
            for (int v = 0; v < 2; ++v) {
                const int k = 4*j + 2*hi + v;
                float g = gate_elem(c0, s0, (ln>>3)&1, (k>>3)&1)
                        * gate_elem(c1, s1, (ln>>2)&1, (k>>2)&1)
                        * gate_elem(c2, s2, (ln>>1)&1, (k>>1)&1)
                        * gate_elem(c3, s3,  ln     &1,  k     &1);
                if (v == 0) a[j].x = g; else a[j].y = g;
            }
        }

        // 32 disjoint 16x16 column tiles per pass; wave w owns tiles {w, w+16}
        #pragma unroll
        for (int tt = 0; tt < 2; ++tt) {
            const int t   = wave + 16*tt;
            const int col = 16*t + ln;

            // B fragments: chunk j holds X[4j..4j+3, cols]; lane l: N=ln, K=4j+v+2*hi
            v2f b[4];
            #pragma unroll
            for (int j = 0; j < 4; ++j) {
                const int k0 = 4*j + 2*hi;
                b[j].x = st[paddr(pass, k0,     col)];
                b[j].y = st[paddr(pass, k0 + 1, col)];
            }

            v8f d = {0.f,0.f,0.f,0.f,0.f,0.f,0.f,0.f};
            #pragma unroll
            for (int j = 0; j < 4; ++j) {
                d = __builtin_amdgcn_wmma_f32_16x16x4_f32(
                        false, a[j], false, b[j], (short)0, d, false, false);
            }

            // D: VGPR v, lane l -> row r = v + 8*hi, col = ln ; store in place
            #pragma unroll
            for (int v = 0; v < 8; ++v) {
                const int r = v + 8*hi;
                st[paddr(pass, r, col)] = d[v];
            }
        }
        __syncthreads();
    }

    // ---- site 12 (bit 0): plain 2x2 rotation on adjacent pairs ----
    {
        const float c = gc[12], s = gs[12];
        #pragma unroll
        for (int i = 0; i < (DIM/2)/NT; ++i) {
            const int p = tid + i*NT;
            float x0 = st[2*p], x1 = st[2*p + 1];
            st[2*p]     =  c*x0 + s*x1;
            st[2*p + 1] = -s*x0 + c*x1;
        }
    }
    __syncthreads();

    // ---- write this component to out[comp*DIM ..] (output is [2, 8192]) ----
    {
        v4f* ov = (v4f*)(out + comp*DIM);
        const v4f* sv = (const v4f*)st;
        #pragma unroll
        for (int i = 0; i < DIM/4/NT; ++i) ov[tid + i*NT] = sv[tid + i*NT];
    }
}

extern "C" void kernel_launch(void* const* d_in, const int* in_sizes, int n_in,
                              void* d_out, int out_size, void* d_ws, size_t ws_size,
                              hipStream_t stream) {
    const float* xr  = (const float*)d_in[0];   // x_real, 8192
    const float* xi  = (const float*)d_in[1];   // x_imag, 8192
    const float* ang = (const float*)d_in[2];   // angle, 13
    float* out = (float*)d_out;                 // [2, 8192] float32

    hipLaunchKernelGGL(ry13_wmma_kernel, dim3(2), dim3(NT), 0, stream,
                       xr, xi, ang, out);
}